// TemporalGCN_32341103739500
// MI455X (gfx1250) — compile-verified
//
#include <hip/hip_runtime.h>
#include <math.h>

typedef __attribute__((ext_vector_type(2))) float v2f;
typedef __attribute__((ext_vector_type(8))) float v8f;

#define WMMA_F32(a, b, c) \
  __builtin_amdgcn_wmma_f32_16x16x4_f32(false, (a), false, (b), (short)0, (c), false, false)

#define RPW 4   // 16-row strips per wave (B fragments register-resident across strips)

// ---------------------------------------------------------------- utility ---

__global__ void k_init(float* __restrict__ dis, float* __restrict__ sums, int n) {
  int i = blockIdx.x * blockDim.x + threadIdx.x;
  if (i < n) dis[i] = 1.0f;          // self-loop contributes 1 to every degree
  if (i < 32) sums[i] = 0.0f;        // pooled-output accumulator
}

__global__ void k_deg(const int* __restrict__ dst, float* __restrict__ dis, int e) {
  int i = blockIdx.x * blockDim.x + threadIdx.x;
  if (i < e) atomicAdd(&dis[dst[i]], 1.0f);
}

__global__ void k_rsqrt(float* __restrict__ dis, int n) {
  int i = blockIdx.x * blockDim.x + threadIdx.x;
  if (i < n) dis[i] = rsqrtf(dis[i]);  // deg >= 1 always (self loops)
}

// -------------------------------------------------- fp32 WMMA GEMM, Ncol=64 ---
// One wave computes RPW 16x64 output strips. All B (=W) fragments are loaded
// ONCE per wave into VGPRs (16 K-steps x 4 col-tiles x v2f = 128 floats/lane)
// and reused across strips. __launch_bounds__(128,1) relaxes the VGPR budget
// so the B array stays register-resident (no scratch spills).
// W row-major [64 x 64]. nrows % 16 == 0.
__global__ void __launch_bounds__(128, 1)
k_gemm64(const float* __restrict__ X, const float* __restrict__ W,
         float* __restrict__ O, int nrows) {
  int wave = blockIdx.x * (blockDim.x >> 5) + (threadIdx.x >> 5);
  int lane = threadIdx.x & 31;
  int strip0 = wave * RPW;
  if ((strip0 << 4) >= nrows) return;        // wave-uniform
  int m  = lane & 15;                        // A row / B,D column within tile
  int kh = (lane >> 4) << 1;                 // lanes 16-31 hold K = k0+2, k0+3

  // B fragments: lane holds cols {m, m+16, m+32, m+48}, K = kk*4+kh, +1
  v2f b[16][4];
#pragma unroll
  for (int kk = 0; kk < 16; ++kk) {
    const float* wb = W + (size_t)(kk * 4 + kh) * 64 + m;
#pragma unroll
    for (int t = 0; t < 4; ++t)
      b[kk][t] = v2f{ wb[16 * t], wb[16 * t + 64] };
  }

  for (int s = 0; s < RPW; ++s) {
    int row0 = (strip0 + s) << 4;
    if (row0 >= nrows) return;               // uniform: strips are monotonic
    const float* xrow = X + (size_t)(row0 + m) * 64;
    __builtin_prefetch(xrow + 16 * 64, 0, 1);   // next strip's A row -> cache
    v8f acc0 = {}, acc1 = {}, acc2 = {}, acc3 = {};
#pragma unroll
    for (int kk = 0; kk < 16; ++kk) {
      v2f a = *(const v2f*)(xrow + kk * 4 + kh);
      acc0 = WMMA_F32(a, b[kk][0], acc0);
      acc1 = WMMA_F32(a, b[kk][1], acc1);
      acc2 = WMMA_F32(a, b[kk][2], acc2);
      acc3 = WMMA_F32(a, b[kk][3], acc3);
    }
    // D layout: lane<16 -> col m, VGPR r = row r; lane>=16 -> row r+8
    int mb = (lane >> 4) << 3;
#pragma unroll
    for (int r = 0; r < 8; ++r) {
      float* orow = O + (size_t)(row0 + mb + r) * 64 + m;
      orow[0]  = acc0[r];
      orow[16] = acc1[r];
      orow[32] = acc2[r];
      orow[48] = acc3[r];
    }
  }
}

// ---------------------------------------------- aggregation (self loop init) ---
__global__ void k_agg_init(const float* __restrict__ h, const float* __restrict__ dis,
                           float* __restrict__ agg, int total) {
  int i = blockIdx.x * blockDim.x + threadIdx.x;
  if (i < total) {
    float d = dis[i >> 6];
    agg[i] = h[i] * d * d;                   // self-loop: norm = dis[i]*dis[i]
  }
}

// One wave32 per edge; each lane moves 2 floats of the 64-wide feature row.
__global__ void k_edge(const int* __restrict__ src, const int* __restrict__ dst,
                       const float* __restrict__ dis, const float* __restrict__ h,
                       float* __restrict__ agg, int e) {
  int wave = blockIdx.x * (blockDim.x >> 5) + (threadIdx.x >> 5);
  int lane = threadIdx.x & 31;
  if (wave >= e) return;
  int s = src[wave];
  int d = dst[wave];
  float norm = dis[s] * dis[d];
  v2f hv = *(const v2f*)(h + (size_t)s * 64 + lane * 2);
  float* ap = agg + (size_t)d * 64 + lane * 2;
  atomicAdd(ap,     hv.x * norm);
  atomicAdd(ap + 1, hv.y * norm);
}

__global__ void k_bias_relu(float* __restrict__ a, const float* __restrict__ b, int total) {
  int i = blockIdx.x * blockDim.x + threadIdx.x;
  if (i < total) a[i] = fmaxf(a[i] + b[i & 63], 0.0f);
}

// ---------------------- head: fused GEMM(64->32) + bias + ReLU + sum-pooling ---
// Wl row-major [64 x 32]. One wave per RPW 16-row strips; B register-resident;
// reduces rows locally then atomically accumulates into sums[32].
__global__ void __launch_bounds__(128, 1)
k_pool(const float* __restrict__ X, const float* __restrict__ Wl,
       const float* __restrict__ bl, float* __restrict__ sums, int nrows) {
  int wave = blockIdx.x * (blockDim.x >> 5) + (threadIdx.x >> 5);
  int lane = threadIdx.x & 31;
  int strip0 = wave * RPW;
  if ((strip0 << 4) >= nrows) return;
  int m  = lane & 15;
  int kh = (lane >> 4) << 1;

  v2f b[16][2];
#pragma unroll
  for (int kk = 0; kk < 16; ++kk) {
    const float* wb = Wl + (size_t)(kk * 4 + kh) * 32 + m;
    b[kk][0] = v2f{ wb[0],  wb[32] };
    b[kk][1] = v2f{ wb[16], wb[48] };
  }
  float b0v = bl[m], b1v = bl[m + 16];
  float s0 = 0.0f, s1 = 0.0f;

  for (int s = 0; s < RPW; ++s) {
    int row0 = (strip0 + s) << 4;
    if (row0 >= nrows) break;                // uniform
    const float* xrow = X + (size_t)(row0 + m) * 64;
    __builtin_prefetch(xrow + 16 * 64, 0, 1);
    v8f acc0 = {}, acc1 = {};
#pragma unroll
    for (int kk = 0; kk < 16; ++kk) {
      v2f a = *(const v2f*)(xrow + kk * 4 + kh);
      acc0 = WMMA_F32(a, b[kk][0], acc0);
      acc1 = WMMA_F32(a, b[kk][1], acc1);
    }
    // lane<16 holds rows 0..7 of col m; lane>=16 holds rows 8..15 of col m.
#pragma unroll
    for (int r = 0; r < 8; ++r) {
      s0 += fmaxf(acc0[r] + b0v, 0.0f);
      s1 += fmaxf(acc1[r] + b1v, 0.0f);
    }
  }
  atomicAdd(&sums[m],      s0);
  atomicAdd(&sums[m + 16], s1);
}

__global__ void k_sigmoid(const float* __restrict__ sums, float* __restrict__ out) {
  int i = threadIdx.x;
  if (i < 32) out[i] = 1.0f / (1.0f + expf(-sums[i]));
}

// --------------------------------------------------------------- dispatcher ---

extern "C" void kernel_launch(void* const* d_in, const int* in_sizes, int n_in,
                              void* d_out, int out_size, void* d_ws, size_t ws_size,
                              hipStream_t stream) {
  const float* x  = (const float*)d_in[0];
  const int*   ei = (const int*)d_in[1];   // [2, E]: row0 = src, row1 = dst
  const float* W1 = (const float*)d_in[3];
  const float* b1 = (const float*)d_in[4];
  const float* W2 = (const float*)d_in[5];
  const float* b2 = (const float*)d_in[6];
  const float* Wl = (const float*)d_in[7];
  const float* bl = (const float*)d_in[8];

  const int N = in_sizes[0] / 64;
  const int E = in_sizes[1] / 2;
  const int* src = ei;
  const int* dst = ei + E;

  float* ws   = (float*)d_ws;
  float* dis  = ws;                          // [N]
  float* sums = ws + N;                      // [32]
  float* bufA = ws + N + 64;                 // [N*64] GEMM output h
  float* bufB = bufA + (size_t)N * 64;       // [N*64] aggregation / activations

  const int total = N * 64;
  const int gElem = (total + 255) / 256;     // elementwise grids
  const int gN    = (N + 255) / 256;
  const int gE    = (E + 255) / 256;
  const int strips = (N + 15) / 16;
  const int waves  = (strips + RPW - 1) / RPW;
  const int gGemm  = (waves + 3) / 4;        // 128 threads = 4 waves / block
  const int gEdge  = (E + 7) / 8;            // 256 threads = 8 waves / block

  // normalization: deg -> rsqrt
  k_init  <<<gN, 256, 0, stream>>>(dis, sums, N);
  k_deg   <<<gE, 256, 0, stream>>>(dst, dis, E);
  k_rsqrt <<<gN, 256, 0, stream>>>(dis, N);

  // layer 1: h = x@W1 ; agg = self + edges ; relu(agg + b1)
  k_gemm64   <<<gGemm, 128, 0, stream>>>(x, W1, bufA, N);
  k_agg_init <<<gElem, 256, 0, stream>>>(bufA, dis, bufB, total);
  k_edge     <<<gEdge, 256, 0, stream>>>(src, dst, dis, bufA, bufB, E);
  k_bias_relu<<<gElem, 256, 0, stream>>>(bufB, b1, total);

  // layer 2 (ping-pong: bufB -> bufA -> bufB)
  k_gemm64   <<<gGemm, 128, 0, stream>>>(bufB, W2, bufA, N);
  k_agg_init <<<gElem, 256, 0, stream>>>(bufA, dis, bufB, total);
  k_edge     <<<gEdge, 256, 0, stream>>>(src, dst, dis, bufA, bufB, E);
  k_bias_relu<<<gElem, 256, 0, stream>>>(bufB, b2, total);

  // head: relu(h@Wl + bl), sum over nodes, sigmoid
  k_pool   <<<gGemm, 128, 0, stream>>>(bufB, Wl, bl, sums, N);
  k_sigmoid<<<1, 32, 0, stream>>>(sums, (float*)d_out);
}